// HeterogeneousGINRegressor_87686052315186
// MI455X (gfx1250) — compile-verified
//
#include <hip/hip_runtime.h>
#include <hip/hip_bf16.h>

typedef __attribute__((ext_vector_type(2))) float v2f;
typedef __attribute__((ext_vector_type(8))) float v8f;

// ---------------------------------------------------------------------------
// Wave-level GEMM: one wave computes a 16x128 output stripe.
//   lA : 16 x K tile, row-major, in LDS (this wave's rows)
//   lWT: 128 x K (i.e. W transposed, lWT[col*K + k]), in LDS (shared)
//   c  : 8 accumulators (8 N-tiles of 16 cols), v8f each (C-frag layout)
// Fragment layouts per CDNA5 ISA 7.12.2 (wave32):
//   A 16x4 : lane(l): row = l&15, k = kk + 2*(l>>4) + {0,1}      -> v2f
//   B 4x16 : lane(l): col = l&15, k = kk + 2*(l>>4) + {0,1}      -> v2f
//   C 16x16: vgpr i : lane(l): row = i + 8*(l>>4), col = l&15
// ---------------------------------------------------------------------------
template <int K>
__device__ __forceinline__ void wave_gemm16xK(const float* lA, const float* lWT, v8f c[8]) {
  const int lane = threadIdx.x & 31;
  const int r = lane & 15;
  const int half = lane >> 4;
  for (int kk = 0; kk < K; kk += 4) {
    const int k0 = kk + half * 2;
    v2f a = *(const v2f*)(lA + r * K + k0);
#pragma unroll
    for (int nt = 0; nt < 8; ++nt) {
      v2f b = *(const v2f*)(lWT + (nt * 16 + r) * K + k0);
      c[nt] = __builtin_amdgcn_wmma_f32_16x16x4_f32(
          /*neg_a=*/false, a, /*neg_b=*/false, b,
          /*c_mod=*/(short)0, c[nt], /*reuse_a=*/false, /*reuse_b=*/false);
    }
  }
}

// ---------------------------------------------------------------------------
// Input projection: H = relu(X @ W + b), X: nrows x 64, W: 64 x 128.
// Block: 256 threads, 128 rows. Dynamic LDS: 64KB (WT 32KB + X 32KB).
// ---------------------------------------------------------------------------
__global__ void inproj_kernel(const float* __restrict__ X, const float* __restrict__ W,
                              const float* __restrict__ bias, float* __restrict__ H,
                              int nrows) {
  extern __shared__ float smem[];
  float* lWT = smem;          // 128 cols x 64 k = 8192 floats
  float* lX  = smem + 8192;   // 128 rows x 64   = 8192 floats
  const int t = threadIdx.x;
  const long row0 = (long)blockIdx.x * 128;

  for (int idx = t; idx < 64 * 128; idx += 256) {
    int rr = idx >> 7, cc = idx & 127;
    lWT[cc * 64 + rr] = W[idx];
  }
  for (int i4 = t; i4 < 128 * 16; i4 += 256) {  // 2048 float4
    int lr = i4 >> 4, c4 = i4 & 15;
    long gr = row0 + lr; if (gr >= nrows) gr = nrows - 1;
    float4 v = *(const float4*)(X + gr * 64 + c4 * 4);
    *(float4*)(lX + lr * 64 + c4 * 4) = v;
  }
  __syncthreads();

  const int w = t >> 5, lane = t & 31, r = lane & 15, half = lane >> 4;
  v8f c[8] = {};
  wave_gemm16xK<64>(lX + (w * 16) * 64, lWT, c);
#pragma unroll
  for (int nt = 0; nt < 8; ++nt) {
    const int col = nt * 16 + r;
    const float bb = bias[col];
#pragma unroll
    for (int i = 0; i < 8; ++i) {
      long row = row0 + w * 16 + i + half * 8;
      if (row < nrows) {
        float v = c[nt][i] + bb;
        H[row * 128 + col] = v > 0.f ? v : 0.f;
      }
    }
  }
}

// ---------------------------------------------------------------------------
// Fused GIN MLP for one edge type:
//   acc += relu((msg + hdst) @ W1 + b1) @ W2 + b2
// Block: 256 threads, 128 rows. Dynamic LDS: 128KB (WT 64KB + Z/T 64KB).
// ---------------------------------------------------------------------------
__global__ void mlp_kernel(const float* __restrict__ msg, const float* __restrict__ hdst,
                           const float* __restrict__ W1, const float* __restrict__ b1,
                           const float* __restrict__ W2, const float* __restrict__ b2,
                           float* __restrict__ acc, int nrows) {
  extern __shared__ float smem[];
  float* lWT = smem;           // 128 x 128 (transposed weights)
  float* lZ  = smem + 16384;   // 128 x 128 (Z, then reused as T)
  const int t = threadIdx.x;
  const long row0 = (long)blockIdx.x * 128;

  for (int idx = t; idx < 16384; idx += 256) {
    int rr = idx >> 7, cc = idx & 127;
    lWT[cc * 128 + rr] = W1[idx];
  }
  for (int i4 = t; i4 < 128 * 32; i4 += 256) {  // 4096 float4
    int lr = i4 >> 5, c4 = i4 & 31;
    long gr = row0 + lr; if (gr >= nrows) gr = nrows - 1;
    const float4 m = *(const float4*)(msg  + gr * 128 + c4 * 4);
    const float4 h = *(const float4*)(hdst + gr * 128 + c4 * 4);
    float4 z; z.x = m.x + h.x; z.y = m.y + h.y; z.z = m.z + h.z; z.w = m.w + h.w;
    *(float4*)(lZ + lr * 128 + c4 * 4) = z;
  }
  __syncthreads();

  const int w = t >> 5, lane = t & 31, r = lane & 15, half = lane >> 4;
  {
    v8f c[8] = {};
    wave_gemm16xK<128>(lZ + (w * 16) * 128, lWT, c);
    // T = relu(c + b1) back into this wave's own 16-row LDS stripe
#pragma unroll
    for (int nt = 0; nt < 8; ++nt) {
      const int col = nt * 16 + r;
      const float bb = b1[col];
#pragma unroll
      for (int i = 0; i < 8; ++i) {
        float v = c[nt][i] + bb;
        lZ[(w * 16 + i + half * 8) * 128 + col] = v > 0.f ? v : 0.f;
      }
    }
  }
  __syncthreads();  // all waves done with W1 + T fully written
  for (int idx = t; idx < 16384; idx += 256) {
    int rr = idx >> 7, cc = idx & 127;
    lWT[cc * 128 + rr] = W2[idx];
  }
  __syncthreads();
  {
    v8f c[8] = {};
    wave_gemm16xK<128>(lZ + (w * 16) * 128, lWT, c);
#pragma unroll
    for (int nt = 0; nt < 8; ++nt) {
      const int col = nt * 16 + r;
      const float bb = b2[col];
#pragma unroll
      for (int i = 0; i < 8; ++i) {
        long row = row0 + w * 16 + i + half * 8;
        if (row < nrows) {
          float* p = acc + row * 128 + col;
          *p += c[nt][i] + bb;  // safe: one owner block per row, kernels stream-ordered
        }
      }
    }
  }
}

// ---------------------------------------------------------------------------
// Segment sum: msg[e1[e]] += hsrc[e0[e]], 128 floats/edge, wave-per-edge.
// ---------------------------------------------------------------------------
__global__ void scatter_kernel(const float* __restrict__ hsrc, const int* __restrict__ ei,
                               float* __restrict__ msg, int E) {
  long gid = (long)blockIdx.x * blockDim.x + threadIdx.x;
  int e = (int)(gid >> 5);
  int lane = (int)(gid & 31);
  if (e >= E) return;
  const int s = ei[e];
  const int d = ei[(long)E + e];
  const float4 v = *(const float4*)(hsrc + (long)s * 128 + lane * 4);
  float* p = msg + (long)d * 128 + lane * 4;
  atomicAdd(p + 0, v.x);
  atomicAdd(p + 1, v.y);
  atomicAdd(p + 2, v.z);
  atomicAdd(p + 3, v.w);
}

__global__ void zero_kernel(float* __restrict__ p, long n4) {
  long i = (long)blockIdx.x * blockDim.x + threadIdx.x;
  const long stride = (long)gridDim.x * blockDim.x;
  const float4 z = {0.f, 0.f, 0.f, 0.f};
  for (; i < n4; i += stride) ((float4*)p)[i] = z;
}

__global__ void relu_copy_kernel(float* __restrict__ dst, const float* __restrict__ src, long n4) {
  long i = (long)blockIdx.x * blockDim.x + threadIdx.x;
  const long stride = (long)gridDim.x * blockDim.x;
  for (; i < n4; i += stride) {
    float4 v = ((const float4*)src)[i];
    v.x = v.x > 0.f ? v.x : 0.f;
    v.y = v.y > 0.f ? v.y : 0.f;
    v.z = v.z > 0.f ? v.z : 0.f;
    v.w = v.w > 0.f ? v.w : 0.f;
    ((float4*)dst)[i] = v;
  }
}

// out[row] = dot(h_product[row], Wo) + bo ; wave-per-row
__global__ void out_kernel(const float* __restrict__ hp, const float* __restrict__ Wo,
                           const float* __restrict__ bo, float* __restrict__ out, int nrows) {
  long gid = (long)blockIdx.x * blockDim.x + threadIdx.x;
  int row = (int)(gid >> 5);
  int lane = (int)(gid & 31);
  if (row >= nrows) return;
  const float4 h = *(const float4*)(hp + (long)row * 128 + lane * 4);
  const float4 wv = *(const float4*)(Wo + lane * 4);
  float p = h.x * wv.x + h.y * wv.y + h.z * wv.z + h.w * wv.w;
  for (int off = 16; off > 0; off >>= 1) p += __shfl_xor(p, off, 32);
  if (lane == 0) out[row] = p + bo[0];
}

// ---------------------------------------------------------------------------
// Input layout (setup_inputs insertion order; params flattened as a JAX pytree,
// dict keys in sorted order):
//   0..4  : x_product, x_plant, x_supplier, x_warehouse, x_customer  (N x 64 f32)
//   5..10 : ei_supplies, ei_produces, ei_stored_at, ei_ships_to, ei_orders,
//           ei_rated_by  (2 x E int32)
//   11..20: in_proj (sorted: customer, plant, product, supplier, warehouse),
//           each (W 64x128, b 128)
//   21..44: layer0 (sorted rels: orders, produces, rated_by, ships_to,
//           stored_at, supplies), each (W1, b1, W2, b2)
//   45..68: layer1, same order
//   69,70 : Wo (128), bo (1)
// Node-type storage index: product=0 plant=1 supplier=2 warehouse=3 customer=4
// ---------------------------------------------------------------------------
extern "C" void kernel_launch(void* const* d_in, const int* in_sizes, int n_in,
                              void* d_out, int out_size, void* d_ws, size_t ws_size,
                              hipStream_t stream) {
  const int N = in_sizes[0] / 64;
  const int E = in_sizes[5] / 2;
  const long NH = (long)N * 128;

  float* ws  = (float*)d_ws;
  float* h0  = ws;            // 5 * NH : per-type hidden states
  float* acc = ws + 5 * NH;   // 5 * NH : HeteroConv sum accumulators
  float* msg = ws + 10 * NH;  // 1 * NH : segment-sum buffer

  const int* ei[6];
  for (int j = 0; j < 6; ++j) ei[j] = (const int*)d_in[5 + j];

  // in_proj sorted-key index for storage order {product,plant,supplier,warehouse,customer}
  const int ip_sorted[5] = {2, 1, 3, 4, 0};
  // per edge type (reference order): src nt, dst nt, sorted-rel index
  const int e_src[6] = {2, 1, 0, 3, 4, 1};
  const int e_dst[6] = {1, 0, 3, 4, 0, 2};
  const int e_rid[6] = {5, 1, 4, 3, 0, 2};

  const int PB = 11;
  const dim3 blk(256);
  const int gemm_blocks = (N + 127) / 128;
  const int zgrid = 2048;

  // 1) input projections + relu
  for (int nt = 0; nt < 5; ++nt) {
    const float* Wp = (const float*)d_in[PB + ip_sorted[nt] * 2 + 0];
    const float* bp = (const float*)d_in[PB + ip_sorted[nt] * 2 + 1];
    inproj_kernel<<<gemm_blocks, blk, 65536, stream>>>(
        (const float*)d_in[nt], Wp, bp, h0 + nt * NH, N);
  }

  // 2) GNN layers
  const long sthreads = (long)E * 32;
  const int sblocks = (int)((sthreads + 255) / 256);
  for (int l = 0; l < 2; ++l) {
    zero_kernel<<<zgrid, blk, 0, stream>>>(acc, (5 * NH) / 4);
    const int lbase = PB + 10 + l * 24;
    for (int eix = 0; eix < 6; ++eix) {
      zero_kernel<<<zgrid, blk, 0, stream>>>(msg, NH / 4);
      scatter_kernel<<<sblocks, blk, 0, stream>>>(h0 + e_src[eix] * NH, ei[eix], msg, E);
      const int pb = lbase + e_rid[eix] * 4;
      mlp_kernel<<<gemm_blocks, blk, 131072, stream>>>(
          msg, h0 + e_dst[eix] * NH,
          (const float*)d_in[pb + 0], (const float*)d_in[pb + 1],
          (const float*)d_in[pb + 2], (const float*)d_in[pb + 3],
          acc + e_dst[eix] * NH, N);
    }
    relu_copy_kernel<<<zgrid, blk, 0, stream>>>(h0, acc, (5 * NH) / 4);
  }

  // 3) output head on product
  const float* Wo = (const float*)d_in[n_in - 2];
  const float* bo = (const float*)d_in[n_in - 1];
  const long othreads = (long)N * 32;
  out_kernel<<<(int)((othreads + 255) / 256), blk, 0, stream>>>(
      h0, Wo, bo, (float*)d_out, N);
}